// EventSpace_4793183503007
// MI455X (gfx1250) — compile-verified
//
#include <hip/hip_runtime.h>
#include <hip/hip_bf16.h>

#define B_  8
#define T_  48
#define D_  64
#define U_  48
#define INV_SQRT2 0.70710678118654752f

typedef float v2f __attribute__((ext_vector_type(2)));
typedef float v4f __attribute__((ext_vector_type(4)));
typedef float v8f __attribute__((ext_vector_type(8)));

__device__ __forceinline__ float fast_tanh(float x) {
#if __has_builtin(__builtin_amdgcn_tanhf)
    return __builtin_amdgcn_tanhf(x);
#else
    float y;
    asm volatile("v_tanh_f32 %0, %1" : "=v"(y) : "v"(x));
    return y;
#endif
}

// ---------------- Kernel A: x = LeakyReLU(inputs * diag(space)) ----------------
__global__ __launch_bounds__(256) void prep_x_kernel(
    const float* __restrict__ inputs, const float* __restrict__ space,
    float* __restrict__ x) {
    int n = blockIdx.x * 256 + threadIdx.x;           // over B*T*D = 24576
    if (n >= B_ * T_ * D_) return;
    int k = n & (D_ - 1);
    int j = (n >> 6) % T_;
    float diag = space[((j * T_ + j) * D_ + k) * D_ + k];
    float v = inputs[n] * diag;
    x[n] = v >= 0.f ? v : 0.2f * v;
}

// ---------------- Kernel B: u_hat = x @ caps_W  (WMMA f32 16x16x4) -------------
// X: (384, 64) row-major; W: (64, 2304) row-major.
// Output scattered directly into (B, U, T, U) layout.
__global__ __launch_bounds__(32) void uhat_wmma_kernel(
    const float* __restrict__ X, const float* __restrict__ W,
    float* __restrict__ UH) {
    const int tileM = blockIdx.x;     // 0..23   (384/16)
    const int tileN = blockIdx.y;     // 0..143  (2304/16)
    const int lane  = threadIdx.x;    // 0..31
    const int half  = lane >> 4;      // 0: K pair {0,1}, 1: K pair {2,3}
    const int l16   = lane & 15;

    const int m0 = tileM * 16;
    const int n0 = tileN * 16;
    v8f acc = {};

    #pragma unroll
    for (int k0 = 0; k0 < D_; k0 += 4) {
        // A 16x4: lane l16 holds row M=m0+l16; vgprs hold K = k0+2*half+{0,1}
        const float* ap = X + (m0 + l16) * D_ + k0 + 2 * half;
        v2f a; a.x = ap[0]; a.y = ap[1];
        // B 4x16: lane l16 holds col N=n0+l16; vgprs hold K = k0+2*half+{0,1}
        const float* bp = W + (k0 + 2 * half) * (U_ * U_) + n0 + l16;
        v2f b; b.x = bp[0]; b.y = bp[U_ * U_];
        acc = __builtin_amdgcn_wmma_f32_16x16x4_f32(
            false, a, false, b, (short)0, acc, false, false);
    }

    // D layout: element r -> M = m0 + r + 8*half, N = n0 + l16
    #pragma unroll
    for (int r = 0; r < 8; ++r) {
        int M = m0 + r + 8 * half;     // = b*T + t   (48 = 3 tiles, no tile crosses batch)
        int N = n0 + l16;              // = u*U + d
        int bb = M / T_;  int t = M % T_;
        int u  = N / U_;  int d = N % U_;
        UH[(((size_t)bb * U_ + u) * T_ + t) * U_ + d] = acc[r];
    }
}

// ---------------- Kernel C: dynamic routing + squash + encoder -> levels -------
__global__ __launch_bounds__(256) void routing_levels_kernel(
    const float* __restrict__ UH, const float* __restrict__ kt,
    const float* __restrict__ kf, const float* __restrict__ encb,
    float* __restrict__ levels_out) {
    __shared__ float c[U_ * T_];    // routing coeffs (u,t); reused as tmp (s,q)
    __shared__ float bb[U_ * T_];   // logits (u,t)
    __shared__ float o[U_ * U_];    // capsule outputs (u,d)

    const int b   = blockIdx.x;
    const int tid = threadIdx.x;
    const float* uh = UH + (size_t)b * U_ * T_ * U_;   // (u,t,d)

    for (int n = tid; n < U_ * T_; n += 256) bb[n] = 0.f;
    __syncthreads();

    for (int it = 0; it < 3; ++it) {
        // softmax over u for each t
        if (tid < T_) {
            int t = tid;
            float mx = -1e30f;
            for (int u = 0; u < U_; ++u) mx = fmaxf(mx, bb[u * T_ + t]);
            float s = 0.f;
            for (int u = 0; u < U_; ++u) {
                float e = __expf(bb[u * T_ + t] - mx);
                c[u * T_ + t] = e; s += e;
            }
            float inv = 1.f / s;
            for (int u = 0; u < U_; ++u) c[u * T_ + t] *= inv;
        }
        __syncthreads();
        // o[u,d] = sum_t c[u,t] * uh[u,t,d]
        for (int n = tid; n < U_ * U_; n += 256) {
            int u = n / U_, d = n % U_;
            float s = 0.f;
            for (int t = 0; t < T_; ++t)
                s += c[u * T_ + t] * uh[((size_t)u * T_ + t) * U_ + d];
            o[n] = s;
        }
        __syncthreads();
        // squash over d
        if (tid < U_) {
            int u = tid;
            float s = 0.f;
            for (int d = 0; d < U_; ++d) { float v = o[u * U_ + d]; s += v * v; }
            float f = (s / (1.f + s)) * __frsqrt_rn(s + 1e-7f);
            for (int d = 0; d < U_; ++d) o[u * U_ + d] *= f;
        }
        __syncthreads();
        if (it < 2) {
            // bb[u,t] += sum_d o[u,d] * uh[u,t,d]
            for (int n = tid; n < U_ * T_; n += 256) {
                int u = n / T_, t = n % T_;
                float s = 0.f;
                for (int d = 0; d < U_; ++d)
                    s += o[u * U_ + d] * uh[((size_t)u * T_ + t) * U_ + d];
                bb[n] += s;
            }
            __syncthreads();
        }
    }

    // tmp[s,q] = sum_p o[p,q] * kt[p,s]   (reuse c)
    for (int n = tid; n < T_ * U_; n += 256) {
        int s = n / U_, q = n % U_;
        float acc = 0.f;
        for (int p = 0; p < U_; ++p) acc += o[p * U_ + q] * kt[p * T_ + s];
        c[s * U_ + q] = acc;
    }
    __syncthreads();
    // levels[s,od] = relu(sum_q tmp[s,q] * kf[q,od] + encb[s,od])
    for (int n = tid; n < T_ * D_; n += 256) {
        int s = n / D_, od = n % D_;
        float acc = 0.f;
        for (int q = 0; q < U_; ++q) acc += c[s * U_ + q] * kf[q * D_ + od];
        acc += encb[n];
        levels_out[(size_t)b * T_ * D_ + n] = fmaxf(acc, 0.f);
    }
}

// ---------------- Kernel D: slice recurrence -> r[0..7] ------------------------
__global__ __launch_bounds__(256) void compute_r_kernel(
    const float* __restrict__ space, const float* __restrict__ levels,
    const float* __restrict__ inputs, float* __restrict__ rout) {
    __shared__ float slice[T_ * T_];
    __shared__ float colsum[T_];
    __shared__ float rsh;
    const int tid = threadIdx.x;

    // slice[i,j] = space[i,j,D-1,D-1]
    for (int n = tid; n < T_ * T_; n += 256)
        slice[n] = space[(size_t)n * D_ * D_ + D_ * D_ - 1];
    __syncthreads();

    for (int b = 0; b < B_; ++b) {
        if (tid < T_) {
            int j = tid; float s = 0.f;
            for (int i = 0; i < T_; ++i) s += slice[i * T_ + j];
            colsum[j] = s;
        }
        __syncthreads();
        if (tid == 0) {
            float m = colsum[0];
            for (int j = 1; j < T_; ++j) m = fmaxf(m, colsum[j]);
            rsh = m; rout[b] = m;
        }
        __syncthreads();
        float r = rsh;
        for (int n = tid; n < T_ * T_; n += 256) {
            int i = n / T_, j = n % T_;
            float lv = levels[b * T_ * D_ + i * D_ + (D_ - 1)];
            float xi = inputs[b * T_ * D_ + j * D_ + (D_ - 1)];
            float t  = fast_tanh(lv * xi * INV_SQRT2);
            slice[n] = slice[n] * (1.f - r) + r * t;
        }
        __syncthreads();
    }
}

// ---------------- Kernel E: parallel scan over (i,j,k,l), 4-wide vectorized ----
// Each thread owns 4 consecutive l positions; b loop in registers; b128 NT stores.
__global__ __launch_bounds__(256) void space_scan_kernel(
    const float* __restrict__ space, const float* __restrict__ levels,
    const float* __restrict__ inputs, const float* __restrict__ rarr,
    float* __restrict__ out_spaces) {
    const int NTOT  = T_ * T_ * D_ * D_;               // 9,437,184
    const int NVEC  = NTOT / 4;                        // 2,359,296
    int v = blockIdx.x * 256 + threadIdx.x;
    if (v >= NVEC) return;
    int n = v * 4;                                     // base element index

    int l  = n & (D_ - 1);                             // multiple of 4
    int k  = (n >> 6) & (D_ - 1);
    int ij = n >> 12;                                  // i*T + j (constant over the 4)
    int j  = ij % T_;
    int i  = ij / T_;

    v4f s = *(const v4f*)(space + n);

    float omr[B_], r[B_], lv[B_];
    v4f xi[B_];
    #pragma unroll
    for (int b = 0; b < B_; ++b) {
        r[b]   = rarr[b];                              // uniform -> scalar loads
        omr[b] = 1.f - r[b];
        lv[b]  = levels[b * T_ * D_ + i * D_ + k];
        xi[b]  = *(const v4f*)(inputs + b * T_ * D_ + j * D_ + l);
    }
    #pragma unroll
    for (int b = 0; b < B_; ++b) {
        v4f t;
        #pragma unroll
        for (int e = 0; e < 4; ++e)
            t[e] = fast_tanh(lv[b] * xi[b][e] * INV_SQRT2);
        #pragma unroll
        for (int e = 0; e < 4; ++e)
            s[e] = s[e] * omr[b] + r[b] * t[e];
        __builtin_nontemporal_store(s, (v4f*)(out_spaces + (size_t)b * NTOT + n));
    }
}

// ---------------- Launcher -----------------------------------------------------
extern "C" void kernel_launch(void* const* d_in, const int* in_sizes, int n_in,
                              void* d_out, int out_size, void* d_ws, size_t ws_size,
                              hipStream_t stream) {
    const float* inputs = (const float*)d_in[0];   // (B,T,D)
    const float* space  = (const float*)d_in[1];   // (T,T,D,D)
    const float* caps_W = (const float*)d_in[2];   // (D, U*U)
    const float* enc_kt = (const float*)d_in[3];   // (U,T)
    const float* enc_kf = (const float*)d_in[4];   // (U,D)
    const float* enc_b  = (const float*)d_in[5];   // (T,D)

    float* out       = (float*)d_out;
    float* levels    = out;                          // first B*T*D floats
    float* spaces    = out + B_ * T_ * D_;           // then (B,T,T,D,D)

    float* ws   = (float*)d_ws;
    float* x    = ws;                                // B*T*D      = 24576
    float* uhat = ws + B_ * T_ * D_;                 // B*U*T*U    = 884736
    float* rarr = uhat + B_ * U_ * T_ * U_;          // 8

    // A: preprocess
    {
        int n = B_ * T_ * D_;
        prep_x_kernel<<<(n + 255) / 256, 256, 0, stream>>>(inputs, space, x);
    }
    // B: u_hat GEMM via WMMA f32 16x16x4
    {
        dim3 grid((B_ * T_) / 16, (U_ * U_) / 16);   // 24 x 144 tiles
        uhat_wmma_kernel<<<grid, 32, 0, stream>>>(x, caps_W, uhat);
    }
    // C: routing + encoder -> levels
    routing_levels_kernel<<<B_, 256, 0, stream>>>(uhat, enc_kt, enc_kf, enc_b, levels);
    // D: r scalars from the (T,T) diagonal slice recurrence
    compute_r_kernel<<<1, 256, 0, stream>>>(space, levels, inputs, rarr);
    // E: big streaming scan (302 MB of b128 NT stores, 8x tanh per element)
    {
        int nvec = (T_ * T_ * D_ * D_) / 4;
        space_scan_kernel<<<(nvec + 255) / 256, 256, 0, stream>>>(space, levels, inputs, rarr, spaces);
    }
}